// RPN_87471303950733
// MI455X (gfx1250) — compile-verified
//
#include <hip/hip_runtime.h>
#include <cmath>
#include <stdint.h>

#define H_    128
#define W_    128
#define CIN   256
#define HW    16384
#define NA    147456      /* HW * 9 anchors */
#define PRE_N 6000
#define POST_N 1000
#define CAND_CAP 8192
#define NMS_T 0.7f
#define MIN_SIZE_ 16.0f
#define CCH   16          /* input-channel chunk staged in LDS */
#define CP    20          /* padded channel stride (floats): conflict-free b64 */

typedef __attribute__((ext_vector_type(2))) float v2f;
typedef __attribute__((ext_vector_type(8))) float v8f;

struct AParams { float bw[9], bh[9], bcx[9], bcy[9]; };

// ---------------------------------------------------------------------------
// Kernel 1: 3x3 conv (SAME) + bias + ReLU via V_WMMA_F32_16X16X4_F32.
// Grid: (128 rows y, 8 n-pair-tiles). Block: 256 threads = 8 waves.
// Wave w: x in [16w,16w+16), TWO n-tiles [32*by,32*by+16) and [..+16,..+32).
// LDS layouts are channel-innermost (stride CP=20 floats) so every WMMA
// fragment is a single aligned, bank-conflict-free ds_load_b64.
// ---------------------------------------------------------------------------
__global__ __launch_bounds__(256)
void conv3x3_wmma(const float* __restrict__ feat, const float* __restrict__ w,
                  const float* __restrict__ bias, float* __restrict__ xrelu) {
  __shared__ __align__(16) float fA[3][130][CP];   // rows y-1..y+1, x in [-1,128], c
  __shared__ __align__(16) float wB[9][32][CP];    // [tap][n][c]
  const int y    = blockIdx.x;
  const int n0   = blockIdx.y * 32;
  const int tid  = threadIdx.x;
  const int wave = tid >> 5;
  const int lane = tid & 31;
  const int lm   = lane & 15;
  const int hi   = lane >> 4;        // 0: K pair {0,1}, 1: K pair {2,3}
  const int x0   = wave * 16;
  const int c2   = 2 * hi;

  v8f acc0 = {}, acc1 = {};
  for (int c0 = 0; c0 < CIN; c0 += CCH) {
    __syncthreads();
    // stage A halo rows for this channel chunk (zero-padded), coalesced on x
    for (int i = tid; i < CCH * 3 * 130; i += 256) {
      int cI = i / 390, rem = i % 390, r = rem / 130, xp = rem % 130;
      int xx = xp - 1, yy = y - 1 + r;
      float v = 0.0f;
      if (yy >= 0 && yy < H_ && xx >= 0 && xx < W_)
        v = feat[(c0 + cI) * HW + yy * W_ + xx];
      fA[r][xp][cI] = v;
    }
    // stage B for 32 output channels: per n, 9*CCH contiguous OIHW floats
    for (int i = tid; i < 32 * 9 * CCH; i += 256) {
      int n = i / (9 * CCH), j = i % (9 * CCH);
      float v = w[(size_t)(n0 + n) * (CIN * 9) + c0 * 9 + j];
      wB[j % 9][n][j / 9] = v;
    }
    __syncthreads();
    for (int cI = 0; cI < CCH; cI += 4) {
#pragma unroll
      for (int e = 0; e < 9; ++e) {
        const int ky = e / 3, kx = e % 3;
        // A frag: M=lm (x), K pair selected by lane half -> one ds_load_b64
        v2f a  = *(const v2f*)&fA[ky][x0 + lm + kx][cI + c2];
        // B frags for the two n-tiles, same K striping
        v2f b0 = *(const v2f*)&wB[e][lm][cI + c2];
        v2f b1 = *(const v2f*)&wB[e][16 + lm][cI + c2];
        acc0 = __builtin_amdgcn_wmma_f32_16x16x4_f32(
            false, a, false, b0, (short)0, acc0, false, false);
        acc1 = __builtin_amdgcn_wmma_f32_16x16x4_f32(
            false, a, false, b1, (short)0, acc1, false, false);
      }
    }
  }
  // epilogue: C layout VGPR v -> M = v + 8*hi, N = lm; 8 consecutive x per lane
  const int xb = x0 + 8 * hi;
  {
    const float bv = bias[n0 + lm];
    float* out = xrelu + (size_t)(n0 + lm) * HW + y * W_ + xb;
    float4 o0, o1;
    o0.x = fmaxf(acc0[0] + bv, 0.0f); o0.y = fmaxf(acc0[1] + bv, 0.0f);
    o0.z = fmaxf(acc0[2] + bv, 0.0f); o0.w = fmaxf(acc0[3] + bv, 0.0f);
    o1.x = fmaxf(acc0[4] + bv, 0.0f); o1.y = fmaxf(acc0[5] + bv, 0.0f);
    o1.z = fmaxf(acc0[6] + bv, 0.0f); o1.w = fmaxf(acc0[7] + bv, 0.0f);
    *(float4*)out = o0; *(float4*)(out + 4) = o1;
  }
  {
    const float bv = bias[n0 + 16 + lm];
    float* out = xrelu + (size_t)(n0 + 16 + lm) * HW + y * W_ + xb;
    float4 o0, o1;
    o0.x = fmaxf(acc1[0] + bv, 0.0f); o0.y = fmaxf(acc1[1] + bv, 0.0f);
    o0.z = fmaxf(acc1[2] + bv, 0.0f); o0.w = fmaxf(acc1[3] + bv, 0.0f);
    o1.x = fmaxf(acc1[4] + bv, 0.0f); o1.y = fmaxf(acc1[5] + bv, 0.0f);
    o1.z = fmaxf(acc1[6] + bv, 0.0f); o1.w = fmaxf(acc1[7] + bv, 0.0f);
    *(float4*)out = o0; *(float4*)(out + 4) = o1;
  }
}

// ---------------------------------------------------------------------------
// Kernel 2: 1x1 heads (cls 18, box 36) + softmax + anchor decode + sort keys
// ---------------------------------------------------------------------------
__global__ __launch_bounds__(256)
void head_decode(const float* __restrict__ xrelu,
                 const float* __restrict__ cls_w, const float* __restrict__ cls_b,
                 const float* __restrict__ box_w, const float* __restrict__ box_b,
                 const int* __restrict__ ih_p, const int* __restrict__ iw_p,
                 AParams ap,
                 float* __restrict__ scores, float* __restrict__ boxes,
                 unsigned long long* __restrict__ keys) {
  __shared__ float wS[54 * 256];
  const int tid = threadIdx.x;
  for (int i = tid; i < 18 * 256; i += 256) wS[i] = cls_w[i];
  for (int i = tid; i < 36 * 256; i += 256) wS[18 * 256 + i] = box_w[i];
  __syncthreads();

  const int pix = blockIdx.x * 256 + tid;
  const int y = pix >> 7, x = pix & 127;

  float acc[54];
#pragma unroll
  for (int o = 0; o < 18; ++o) acc[o] = cls_b[o];
#pragma unroll
  for (int o = 0; o < 36; ++o) acc[18 + o] = box_b[o];

  for (int c = 0; c < 256; ++c) {
    const float f = xrelu[(size_t)c * HW + pix];
#pragma unroll
    for (int o = 0; o < 18; ++o) acc[o] += f * wS[o * 256 + c];
#pragma unroll
    for (int o = 0; o < 36; ++o) acc[18 + o] += f * wS[(18 + o) * 256 + c];
  }

  // softmax over 18 channels
  float m = acc[0];
#pragma unroll
  for (int o = 1; o < 18; ++o) m = fmaxf(m, acc[o]);
  float se = 0.0f;
#pragma unroll
  for (int o = 0; o < 18; ++o) se += expf(acc[o] - m);
  const float inv_s = 1.0f / se;

  const float iw  = (float)iw_p[0];
  const float ihh = (float)ih_p[0];
  const float sx = x * 32.0f, sy = y * 32.0f;

#pragma unroll
  for (int a = 0; a < 9; ++a) {
    const float wa = ap.bw[a], ha = ap.bh[a];
    const float cxa = ap.bcx[a] + sx, cya = ap.bcy[a] + sy;
    const float d0 = acc[18 + 4 * a + 0], d1 = acc[18 + 4 * a + 1];
    const float d2 = acc[18 + 4 * a + 2], d3 = acc[18 + 4 * a + 3];
    const float cx = d0 * wa + cxa;
    const float cy = d1 * ha + cya;
    const float pw = wa * expf(d2);
    const float ph = ha * expf(d3);
    const float x1 = fminf(fmaxf(cx - 0.5f * pw, 0.0f), iw - 1.0f);
    const float y1 = fminf(fmaxf(cy - 0.5f * ph, 0.0f), ihh - 1.0f);
    const float x2 = fminf(fmaxf(cx + 0.5f * pw, 0.0f), iw - 1.0f);
    const float y2 = fminf(fmaxf(cy + 0.5f * ph, 0.0f), ihh - 1.0f);
    const bool valid = (x2 - x1 + 1.0f >= MIN_SIZE_) && (y2 - y1 + 1.0f >= MIN_SIZE_);
    const float pr = expf(acc[9 + a] - m) * inv_s;
    const float sc = valid ? pr : -INFINITY;
    const int i = pix * 9 + a;
    scores[i] = sc;
    boxes[4 * i + 0] = x1; boxes[4 * i + 1] = y1;
    boxes[4 * i + 2] = x2; boxes[4 * i + 3] = y2;
    // sortable key: ascending u64  ==  (score desc, index asc)
    unsigned u = __float_as_uint(sc);
    unsigned mono = (u & 0x80000000u) ? ~u : (u | 0x80000000u);
    unsigned kh = ~mono;
    keys[i] = ((unsigned long long)kh << 32) | (unsigned)i;
  }
}

// ---------------------------------------------------------------------------
// Kernel 3: one round of device-side 8-bit radix select on key hi32.
// state[0]=prefix, state[1]=remaining, state[2]=threshold T, state[3]=cand cnt
// ---------------------------------------------------------------------------
__global__ __launch_bounds__(1024)
void radix_round(const unsigned long long* __restrict__ keys,
                 unsigned* __restrict__ state, int round) {
  __shared__ unsigned hist[256];
  const int tid = threadIdx.x;
  if (tid < 256) hist[tid] = 0;
  __syncthreads();
  const unsigned prefix   = (round == 0) ? 0u : state[0];
  unsigned       target   = (round == 0) ? (unsigned)PRE_N : state[1];
  const unsigned maskHigh = (round == 0) ? 0u : (0xFFFFFFFFu << (32 - 8 * round));
  const int shift = 24 - 8 * round;
  for (int i = tid; i < NA; i += 1024) {
    unsigned h = (unsigned)(keys[i] >> 32);
    if ((h & maskHigh) == prefix)
      atomicAdd(&hist[(h >> shift) & 0xFFu], 1u);
  }
  __syncthreads();
  if (tid == 0) {
    unsigned cum = 0, chosen = 255u;
    for (int b = 0; b < 256; ++b) {
      unsigned c = hist[b];
      if (cum + c >= target) { chosen = (unsigned)b; target -= cum; break; }
      cum += c;
    }
    state[0] = prefix | (chosen << shift);
    state[1] = target;
    if (round == 3) { state[2] = prefix | (chosen << shift); state[3] = 0u; }
  }
}

// ---------------------------------------------------------------------------
// Kernel 4: gather all keys with hi32 <= T into candidate buffer
// ---------------------------------------------------------------------------
__global__ void compact_cand(const unsigned long long* __restrict__ keys,
                             unsigned* __restrict__ state,
                             unsigned long long* __restrict__ cand) {
  const int i = blockIdx.x * 256 + threadIdx.x;
  if (i >= NA) return;
  const unsigned long long k = keys[i];
  if ((unsigned)(k >> 32) <= state[2]) {
    unsigned pos = atomicAdd(&state[3], 1u);
    if (pos < CAND_CAP) cand[pos] = k;
  }
}

// ---------------------------------------------------------------------------
// Kernel 5: single-block bitonic sort of candidates -> sorted top-6000
// ---------------------------------------------------------------------------
__global__ __launch_bounds__(1024)
void sort_topk(const unsigned long long* __restrict__ cand,
               const unsigned* __restrict__ state,
               const float* __restrict__ boxes,
               float* __restrict__ tscores, float* __restrict__ tboxes) {
  __shared__ unsigned long long sm[CAND_CAP];   // 64 KB
  const int tid = threadIdx.x;
  unsigned cnt = state[3];
  if (cnt > CAND_CAP) cnt = CAND_CAP;
  for (int i = tid; i < CAND_CAP; i += 1024)
    sm[i] = (i < (int)cnt) ? cand[i] : 0xFFFFFFFFFFFFFFFFull;
  __syncthreads();
  for (int k = 2; k <= CAND_CAP; k <<= 1) {
    for (int j = k >> 1; j > 0; j >>= 1) {
      for (int i = tid; i < CAND_CAP; i += 1024) {
        const int ixj = i ^ j;
        if (ixj > i) {
          unsigned long long A = sm[i], B = sm[ixj];
          const bool up = ((i & k) == 0);
          if ((A > B) == up) { sm[i] = B; sm[ixj] = A; }
        }
      }
      __syncthreads();
    }
  }
  for (int t = tid; t < PRE_N; t += 1024) {
    const unsigned long long kk = sm[t];
    const unsigned idx = (unsigned)(kk & 0xFFFFFFFFull);
    const unsigned khi = (unsigned)(kk >> 32);
    float sc = -INFINITY, b0 = 0, b1 = 0, b2 = 0, b3 = 0;
    if (idx < NA) {
      const unsigned mono = ~khi;
      const unsigned u = (mono & 0x80000000u) ? (mono & 0x7FFFFFFFu) : ~mono;
      sc = __uint_as_float(u);
      b0 = boxes[4 * idx + 0]; b1 = boxes[4 * idx + 1];
      b2 = boxes[4 * idx + 2]; b3 = boxes[4 * idx + 3];
    }
    tscores[t] = sc;
    tboxes[4 * t + 0] = b0; tboxes[4 * t + 1] = b1;
    tboxes[4 * t + 2] = b2; tboxes[4 * t + 3] = b3;
  }
}

// ---------------------------------------------------------------------------
// Kernel 6: sequential NMS (1000 picks) in one block, boxes in registers
// ---------------------------------------------------------------------------
#define SLOTS 6
__global__ __launch_bounds__(1024)
void nms_kernel(const float* __restrict__ tscores, const float* __restrict__ tboxes,
                float* __restrict__ rois) {
  __shared__ float rs[1024];
  __shared__ int   ri[1024];
  __shared__ float bc[8];
  const int t = threadIdx.x;
  float X1[SLOTS], Y1[SLOTS], X2[SLOTS], Y2[SLOTS], AR[SLOTS], SC[SLOTS];
  unsigned aliveM = 0;
#pragma unroll
  for (int q = 0; q < SLOTS; ++q) {
    const int i = q * 1024 + t;
    if (i < PRE_N) {
      const float s = tscores[i];
      X1[q] = tboxes[4 * i + 0]; Y1[q] = tboxes[4 * i + 1];
      X2[q] = tboxes[4 * i + 2]; Y2[q] = tboxes[4 * i + 3];
      AR[q] = (X2[q] - X1[q] + 1.0f) * (Y2[q] - Y1[q] + 1.0f);
      SC[q] = s;
      if (s > -INFINITY) aliveM |= (1u << q);
    } else { SC[q] = -INFINITY; X1[q] = Y1[q] = X2[q] = Y2[q] = 0.0f; AR[q] = 1.0f; }
  }
  for (int it = 0; it < POST_N; ++it) {
    // local argmax (first-index-wins on ties via ascending scan)
    float bs = (aliveM & 1u) ? SC[0] : -INFINITY;
    int   bi = t;
#pragma unroll
    for (int q = 1; q < SLOTS; ++q) {
      const int i = q * 1024 + t;
      if (i < PRE_N) {
        const float v = ((aliveM >> q) & 1u) ? SC[q] : -INFINITY;
        if (v > bs) { bs = v; bi = i; }
      }
    }
    rs[t] = bs; ri[t] = bi;
    __syncthreads();
    for (int off = 512; off > 0; off >>= 1) {
      if (t < off) {
        const float s2 = rs[t + off]; const int i2 = ri[t + off];
        if (s2 > rs[t] || (s2 == rs[t] && i2 < ri[t])) { rs[t] = s2; ri[t] = i2; }
      }
      __syncthreads();
    }
    const int j = ri[0];
    const float sj = rs[0];
    if (t == (j & 1023)) {
      const int q = j >> 10;
      bc[0] = X1[q]; bc[1] = Y1[q]; bc[2] = X2[q]; bc[3] = Y2[q]; bc[4] = AR[q];
    }
    __syncthreads();
    const bool ok = sj > -INFINITY;
    if (t == 0) {
      rois[it * 5 + 0] = 0.0f;
      rois[it * 5 + 1] = ok ? bc[0] : 0.0f;
      rois[it * 5 + 2] = ok ? bc[1] : 0.0f;
      rois[it * 5 + 3] = ok ? bc[2] : 0.0f;
      rois[it * 5 + 4] = ok ? bc[3] : 0.0f;
    }
    const float jx1 = bc[0], jy1 = bc[1], jx2 = bc[2], jy2 = bc[3], ja = bc[4];
#pragma unroll
    for (int q = 0; q < SLOTS; ++q) {
      const int i = q * 1024 + t;
      if (i < PRE_N && ((aliveM >> q) & 1u)) {
        const float xx1 = fmaxf(X1[q], jx1), yy1 = fmaxf(Y1[q], jy1);
        const float xx2 = fminf(X2[q], jx2), yy2 = fminf(Y2[q], jy2);
        const float iw  = fmaxf(xx2 - xx1 + 1.0f, 0.0f);
        const float ih  = fmaxf(yy2 - yy1 + 1.0f, 0.0f);
        const float inter = iw * ih;
        const float iou = inter / (AR[q] + ja - inter);
        if (!(iou <= NMS_T)) aliveM &= ~(1u << q);
      }
    }
    __syncthreads();
  }
}

// ---------------------------------------------------------------------------
extern "C" void kernel_launch(void* const* d_in, const int* in_sizes, int n_in,
                              void* d_out, int out_size, void* d_ws, size_t ws_size,
                              hipStream_t stream) {
  (void)in_sizes; (void)n_in; (void)out_size; (void)ws_size;
  const float* feat  = (const float*)d_in[0];
  const float* rpn_w = (const float*)d_in[1];
  const float* rpn_b = (const float*)d_in[2];
  const float* cls_w = (const float*)d_in[3];
  const float* cls_b = (const float*)d_in[4];
  const float* box_w = (const float*)d_in[5];
  const float* box_b = (const float*)d_in[6];
  const int*   ih    = (const int*)d_in[7];
  const int*   iw    = (const int*)d_in[8];

  char* ws = (char*)d_ws;
  size_t off = 0;
  auto alloc = [&](size_t bytes) -> void* {
    void* p = ws + off;
    off = (off + bytes + 255) & ~((size_t)255);
    return p;
  };
  float* xrelu   = (float*)alloc((size_t)CIN * HW * 4);            // 16.8 MB
  float* scores  = (float*)alloc((size_t)NA * 4);
  float* boxes   = (float*)alloc((size_t)NA * 16);
  unsigned long long* keys = (unsigned long long*)alloc((size_t)NA * 8);
  unsigned long long* cand = (unsigned long long*)alloc((size_t)CAND_CAP * 8);
  float* tscores = (float*)alloc((size_t)PRE_N * 4);
  float* tboxes  = (float*)alloc((size_t)PRE_N * 16);
  unsigned* state = (unsigned*)alloc(64);

  // base anchors (float64 like numpy, nearbyint == np.round half-to-even)
  AParams ap;
  {
    const double cx = 7.5, cy = 7.5, bb = 16.0;
    const double ratios[3] = {0.5, 1.0, 2.0};
    const int scales[3] = {16, 32, 64};
    int idx = 0;
    for (int r = 0; r < 3; ++r) {
      const double wsz = nearbyint(sqrt(bb * bb / ratios[r]));
      const double hsz = nearbyint(wsz * ratios[r]);
      for (int s = 0; s < 3; ++s) {
        const double Wb = wsz * scales[s], Hb = hsz * scales[s];
        const float x1 = (float)(cx - 0.5 * (Wb - 1.0));
        const float y1 = (float)(cy - 0.5 * (Hb - 1.0));
        const float x2 = (float)(cx + 0.5 * (Wb - 1.0));
        const float y2 = (float)(cy + 0.5 * (Hb - 1.0));
        const float bw = x2 - x1 + 1.0f, bh = y2 - y1 + 1.0f;
        ap.bw[idx] = bw; ap.bh[idx] = bh;
        ap.bcx[idx] = x1 + 0.5f * (bw - 1.0f);
        ap.bcy[idx] = y1 + 0.5f * (bh - 1.0f);
        ++idx;
      }
    }
  }

  conv3x3_wmma<<<dim3(128, 8), 256, 0, stream>>>(feat, rpn_w, rpn_b, xrelu);
  head_decode<<<dim3(64), 256, 0, stream>>>(xrelu, cls_w, cls_b, box_w, box_b,
                                            ih, iw, ap, scores, boxes, keys);
  for (int r = 0; r < 4; ++r)
    radix_round<<<1, 1024, 0, stream>>>(keys, state, r);
  compact_cand<<<dim3((NA + 255) / 256), 256, 0, stream>>>(keys, state, cand);
  sort_topk<<<1, 1024, 0, stream>>>(cand, state, boxes, tscores, tboxes);
  nms_kernel<<<1, 1024, 0, stream>>>(tscores, tboxes, (float*)d_out);
}